// GMAELoss_61727269978573
// MI455X (gfx1250) — compile-verified
//
#include <hip/hip_runtime.h>
#include <hip/hip_bf16.h>
#include <math.h>

// Problem sizes from the reference setup_inputs()
static constexpr int Bc  = 2;     // batch
static constexpr int Kc  = 1024;  // queries per batch
static constexpr int Nc  = 8192;  // gaussians per set (N == M)
static constexpr int NCH = 8;     // N-chunks per (ktile, set, b) for parallelism
static constexpr int NF  = 12;    // 10 features padded to 12 (3 WMMA K=4 chunks)
static constexpr int NP  = 6;     // NF/2 feature pairs

typedef __attribute__((ext_vector_type(2))) float v2f;
typedef __attribute__((ext_vector_type(8))) float v8f;

// ---------------------------------------------------------------------------
// Phase 1a: query features  P_k = [qx2,qy2,qz2,qxqy,qxqz,qyqz,qx,qy,qz,1,0,0]
// Layout: Pfeat[(b*K + k)*12 + f]  (8B-aligned pairs for v2f A-fragment loads)
// ---------------------------------------------------------------------------
__global__ void query_feat_kernel(const float* __restrict__ q,
                                  float* __restrict__ Pfeat) {
    int idx = blockIdx.x * blockDim.x + threadIdx.x;   // b*K + k
    if (idx >= Bc * Kc) return;
    float qx = fminf(fmaxf(q[idx * 3 + 0], -1.0f), 1.0f);
    float qy = fminf(fmaxf(q[idx * 3 + 1], -1.0f), 1.0f);
    float qz = fminf(fmaxf(q[idx * 3 + 2], -1.0f), 1.0f);
    float* r = Pfeat + (size_t)idx * NF;
    r[0] = qx * qx; r[1] = qy * qy; r[2] = qz * qz;
    r[3] = qx * qy; r[4] = qx * qz; r[5] = qy * qz;
    r[6] = qx;      r[7] = qy;      r[8] = qz;
    r[9] = 1.0f;    r[10] = 0.0f;   r[11] = 0.0f;
}

// ---------------------------------------------------------------------------
// Phase 1b: per-gaussian features, packed as feature-PAIRS:
//   Gpair[sb][p][n] = (feat 2p, feat 2p+1)  -> one b64 load per B-frag register
//   A = R^T diag(1/(s+1e-8)^2) R ;  b = -2 A x ;  c = x^T A x
// ---------------------------------------------------------------------------
__global__ void gauss_feat_kernel(const float* __restrict__ xyz,
                                  const float* __restrict__ rot,
                                  const float* __restrict__ slog,
                                  const float* __restrict__ olog,
                                  v2f* __restrict__ Gpair,
                                  float* __restrict__ Opac,
                                  int setIdx) {
    int idx = blockIdx.x * blockDim.x + threadIdx.x;   // b*N + n
    if (idx >= Bc * Nc) return;
    int b = idx / Nc, n = idx % Nc;

    // scale -> inverse scale
    float s0 = fminf(fmaxf(__expf(slog[idx * 3 + 0]), 1e-6f), 100.0f);
    float s1 = fminf(fmaxf(__expf(slog[idx * 3 + 1]), 1e-6f), 100.0f);
    float s2 = fminf(fmaxf(__expf(slog[idx * 3 + 2]), 1e-6f), 100.0f);
    float i0 = 1.0f / (s0 + 1e-8f);
    float i1 = 1.0f / (s1 + 1e-8f);
    float i2 = 1.0f / (s2 + 1e-8f);
    float w0 = i0 * i0, w1 = i1 * i1, w2 = i2 * i2;

    // normalized quaternion -> rotation matrix rows
    float qw = rot[idx * 4 + 0], qx = rot[idx * 4 + 1];
    float qy = rot[idx * 4 + 2], qz = rot[idx * 4 + 3];
    float nrm = fmaxf(sqrtf(qw * qw + qx * qx + qy * qy + qz * qz), 1e-12f);
    float inv = 1.0f / nrm;
    qw *= inv; qx *= inv; qy *= inv; qz *= inv;
    float xx = qx * qx, yy = qy * qy, zz = qz * qz;
    float xy = qx * qy, xz = qx * qz, yz = qy * qz;
    float wx = qw * qx, wy = qw * qy, wz = qw * qz;
    float R00 = 1.0f - 2.0f * (yy + zz), R01 = 2.0f * (xy - wz), R02 = 2.0f * (xz + wy);
    float R10 = 2.0f * (xy + wz), R11 = 1.0f - 2.0f * (xx + zz), R12 = 2.0f * (yz - wx);
    float R20 = 2.0f * (xz - wy), R21 = 2.0f * (yz + wx), R22 = 1.0f - 2.0f * (xx + yy);

    // A = R^T diag(w) R (symmetric)
    float A00 = w0 * R00 * R00 + w1 * R10 * R10 + w2 * R20 * R20;
    float A01 = w0 * R00 * R01 + w1 * R10 * R11 + w2 * R20 * R21;
    float A02 = w0 * R00 * R02 + w1 * R10 * R12 + w2 * R20 * R22;
    float A11 = w0 * R01 * R01 + w1 * R11 * R11 + w2 * R21 * R21;
    float A12 = w0 * R01 * R02 + w1 * R11 * R12 + w2 * R21 * R22;
    float A22 = w0 * R02 * R02 + w1 * R12 * R12 + w2 * R22 * R22;

    float px = xyz[idx * 3 + 0], py = xyz[idx * 3 + 1], pz = xyz[idx * 3 + 2];
    float bx = -2.0f * (A00 * px + A01 * py + A02 * pz);
    float by = -2.0f * (A01 * px + A11 * py + A12 * pz);
    float bz = -2.0f * (A02 * px + A12 * py + A22 * pz);
    float c  = -0.5f * (bx * px + by * py + bz * pz);   // x^T A x

    int sb = setIdx * Bc + b;
    v2f* g = Gpair + (size_t)sb * NP * Nc + n;
    g[0 * Nc] = (v2f){A00,        A11};
    g[1 * Nc] = (v2f){A22,        2.0f * A01};
    g[2 * Nc] = (v2f){2.0f * A02, 2.0f * A12};
    g[3 * Nc] = (v2f){bx,         by};
    g[4 * Nc] = (v2f){bz,         c};
    g[5 * Nc] = (v2f){0.0f,       0.0f};

    float o = olog[idx];
    Opac[(size_t)sb * Nc + n] = fminf(fmaxf(1.0f / (1.0f + __expf(-o)), 1e-6f), 1.0f);
}

// ---------------------------------------------------------------------------
// Phase 2: main kernel.  One wave32 per (16-query tile, N-chunk, set*B).
// dist2 tile [16x16] = 3 chained V_WMMA_F32_16X16X4_F32, then w*exp2(max(...)),
// accumulate, butterfly-reduce over the 16-lane halves, atomicAdd density.
// A-frag (ISA 7.12.2, 32-bit A 16x4): lane l row M=l%16; VGPR0 holds
// K = 2*(l/16), VGPR1 holds K+1.  B-frag mirrored (col = l%16).
// ---------------------------------------------------------------------------
__global__ void density_kernel(const float* __restrict__ Pfeat,
                               const v2f* __restrict__ Gpair,
                               const float* __restrict__ Opac,
                               float* __restrict__ Dens) {
    const int lane   = threadIdx.x;        // 0..31
    const int ktile  = blockIdx.x;         // 0..K/16-1
    const int nchunk = blockIdx.y;         // 0..NCH-1
    const int sb     = blockIdx.z;         // set*B + b, 0..3
    const int b      = sb & (Bc - 1);
    const int half   = lane >> 4;          // 0 for lanes 0-15, 1 for 16-31

    const int krow = ktile * 16 + (lane & 15);
    // A fragment: 2 contiguous floats at pair offset (chunk*2 + half)
    const v2f* prow = (const v2f*)(Pfeat + ((size_t)b * Kc + krow) * NF);
    v2f a0 = prow[0 + half];   // feats {0,1} or {2,3}
    v2f a1 = prow[2 + half];   // feats {4,5} or {6,7}
    v2f a2 = prow[4 + half];   // feats {8,9} or {10,11}

    const v2f*   gbase = Gpair + (size_t)sb * NP * Nc;
    const float* obase = Opac + (size_t)sb * Nc;

    float acc[8];
#pragma unroll
    for (int i = 0; i < 8; ++i) acc[i] = 0.0f;

    const int span = Nc / NCH;
    const int n0 = nchunk * span;
    const int n1 = n0 + span;
    const float kExp2  = -0.72134752044448170f;   // -0.5 / ln(2)
    const float kFloor = 50.0f * kExp2;           // clamp dist2<=50 in exp domain

    for (int n = n0; n < n1; n += 16) {
        const int col = n + (lane & 15);
        // B fragment: pair index = chunk*2 + half, one b64 load each
        v2f g0 = gbase[(size_t)(0 + half) * Nc + col];
        v2f g1 = gbase[(size_t)(2 + half) * Nc + col];
        v2f g2 = gbase[(size_t)(4 + half) * Nc + col];
        float w = obase[col];

        v8f ct = {};
        ct = __builtin_amdgcn_wmma_f32_16x16x4_f32(false, a0, false, g0, (short)0, ct, false, false);
        ct = __builtin_amdgcn_wmma_f32_16x16x4_f32(false, a1, false, g1, (short)0, ct, false, false);
        ct = __builtin_amdgcn_wmma_f32_16x16x4_f32(false, a2, false, g2, (short)0, ct, false, false);

#pragma unroll
        for (int i = 0; i < 8; ++i) {
            // arg in [kFloor, 0] -> raw v_exp_f32, no range fixup needed
            float arg = fmaxf(ct[i] * kExp2, kFloor);
            acc[i] = fmaf(w, __builtin_amdgcn_exp2f(arg), acc[i]);
        }
    }

    // Each VGPR i: lanes 0-15 hold (row i, col=lane), lanes 16-31 (row i+8).
    // Sum the 16 columns within each half-wave, then one atomicAdd per row.
#pragma unroll
    for (int i = 0; i < 8; ++i) {
        float v = acc[i];
        v += __shfl_xor(v, 1, 32);
        v += __shfl_xor(v, 2, 32);
        v += __shfl_xor(v, 4, 32);
        v += __shfl_xor(v, 8, 32);
        if ((lane & 15) == 0) {
            int row = ktile * 16 + i + (half << 3);
            atomicAdd(&Dens[(size_t)sb * Kc + row], v);
        }
    }
}

// ---------------------------------------------------------------------------
// Phase 3: KL divergence.  One block of K=1024 threads handles both batches.
// ---------------------------------------------------------------------------
__device__ float block_reduce_sum(float v, float* red) {
    int t = threadIdx.x;
    red[t] = v;
    __syncthreads();
    for (int s = 512; s > 0; s >>= 1) {
        if (t < s) red[t] += red[t + s];
        __syncthreads();
    }
    float r = red[0];
    __syncthreads();
    return r;
}

__global__ void kl_kernel(const float* __restrict__ Dens, float* __restrict__ out) {
    __shared__ float red[1024];
    int k = threadIdx.x;
    float kl_acc = 0.0f;
    for (int b = 0; b < Bc; ++b) {
        float din  = Dens[(size_t)(0 * Bc + b) * Kc + k] + 1e-8f;
        float dout = Dens[(size_t)(1 * Bc + b) * Kc + k] + 1e-8f;
        float sin_ = fmaxf(block_reduce_sum(din, red), 1e-8f);
        float sout = fmaxf(block_reduce_sum(dout, red), 1e-8f);
        float pin  = fmaxf(din / sin_, 1e-8f);
        float pout = fmaxf(dout / sout, 1e-8f);
        float kld  = block_reduce_sum(pin * logf(pin / pout), red);
        kl_acc += isfinite(kld) ? kld : 0.0f;
    }
    if (k == 0) out[0] = kl_acc / (float)Bc;
}

// ---------------------------------------------------------------------------
extern "C" void kernel_launch(void* const* d_in, const int* in_sizes, int n_in,
                              void* d_out, int out_size, void* d_ws, size_t ws_size,
                              hipStream_t stream) {
    const float* queries = (const float*)d_in[0];
    const float* in_xyz  = (const float*)d_in[1];
    const float* in_rot  = (const float*)d_in[2];
    const float* in_scl  = (const float*)d_in[3];
    const float* in_opa  = (const float*)d_in[4];
    const float* ou_xyz  = (const float*)d_in[5];
    const float* ou_rot  = (const float*)d_in[6];
    const float* ou_scl  = (const float*)d_in[7];
    const float* ou_opa  = (const float*)d_in[8];

    float* ws    = (float*)d_ws;
    float* Pfeat = ws;                                  // B*K*12       = 24576 f
    v2f*   Gpair = (v2f*)(Pfeat + (size_t)Bc * Kc * NF);// 2*B*6*N v2f  = 393216 f
    float* Opac  = (float*)(Gpair + (size_t)2 * Bc * NP * Nc); // 2*B*N = 32768 f
    float* Dens  = Opac + (size_t)2 * Bc * Nc;          // 2*B*K        = 4096 f

    hipMemsetAsync(Dens, 0, (size_t)2 * Bc * Kc * sizeof(float), stream);

    query_feat_kernel<<<(Bc * Kc + 255) / 256, 256, 0, stream>>>(queries, Pfeat);
    gauss_feat_kernel<<<(Bc * Nc + 255) / 256, 256, 0, stream>>>(
        in_xyz, in_rot, in_scl, in_opa, Gpair, Opac, 0);
    gauss_feat_kernel<<<(Bc * Nc + 255) / 256, 256, 0, stream>>>(
        ou_xyz, ou_rot, ou_scl, ou_opa, Gpair, Opac, 1);

    dim3 grid(Kc / 16, NCH, 2 * Bc);
    density_kernel<<<grid, 32, 0, stream>>>(Pfeat, Gpair, Opac, Dens);

    kl_kernel<<<1, Kc, 0, stream>>>(Dens, (float*)d_out);
}